// LoRAEmbed_78520592105994
// MI455X (gfx1250) — compile-verified
//
#include <hip/hip_runtime.h>

// LoRA embed: out[m, d] = sum_r lora_down[ids[m], r] * lora_up[r, d]
// M = B*S = 8192 tokens, RANK = 16, DIM = 4096, f32 throughout.
// Store-BW bound (134 MB out); WMMA f32 16x16x4 chained x4 for K=16.

typedef __attribute__((ext_vector_type(2))) float v2f;
typedef __attribute__((ext_vector_type(8))) float v8f;

#define LORA_DIM   4096
#define LORA_RANK  16
#define NT_PER_WAVE 4            // 16-wide N tiles per wave
#define BLOCK_COLS (8 * NT_PER_WAVE * 16)   // 512 columns per block

__global__ __launch_bounds__(256)
void lora_embed_wmma_kernel(const int*   __restrict__ ids,   // [M]
                            const float* __restrict__ down,  // [VOCAB, RANK]
                            const float* __restrict__ up,    // [RANK, DIM]
                            float*       __restrict__ out)   // [M, DIM]
{
    __shared__ float lds_a[16 * LORA_RANK];   // 16 tokens x rank-16 A tile

    const int tid  = threadIdx.x;     // 0..255
    const int wave = tid >> 5;        // 0..7
    const int lane = tid & 31;

    const int m0 = blockIdx.y * 16;               // token tile base
    const int c0 = blockIdx.x * BLOCK_COLS;       // column base of block

    // ---- Stage A tile: gather 16 token rows (rank 16) into LDS ----
    {
        const int tok = tid >> 4;                  // 0..15
        const int r   = tid & 15;                  // 0..15
        const int id  = ids[m0 + tok];
        lds_a[tok * LORA_RANK + r] = down[id * LORA_RANK + r];
    }
    __syncthreads();

    // ---- Hoist A fragments (invariant across N tiles) ----
    // WMMA_F32_16X16X4_F32 A layout: v0 = A[m][k0 + kb], v1 = A[m][k0 + kb + 1]
    // with m = lane&15, kb = (lane<16) ? 0 : 2, k0 = 4*i for chunk i.
    const int rowm = lane & 15;
    const int kb   = (lane >> 4) << 1;            // 0 or 2
    v2f afrag[4];
#pragma unroll
    for (int i = 0; i < 4; ++i) {
        const float* p = &lds_a[rowm * LORA_RANK + i * 4 + kb];
        afrag[i] = *(const v2f*)p;                // 8B-aligned ds_load_b64
    }

    // ---- N-tile loop: 4 WMMAs (K = 16) per 16x16 output tile ----
#pragma unroll
    for (int t = 0; t < NT_PER_WAVE; ++t) {
        const int col = c0 + (wave * NT_PER_WAVE + t) * 16 + rowm;

        // prefetch next tile's B columns into cache hierarchy
        if (t + 1 < NT_PER_WAVE) {
            __builtin_prefetch(&up[kb * LORA_DIM + col + 16], 0, 3);
        }

        v8f c = {};
#pragma unroll
        for (int i = 0; i < 4; ++i) {
            const int k = i * 4 + kb;
            v2f b;
            b.x = up[(k    ) * LORA_DIM + col];   // B v0: K = k0 + kb
            b.y = up[(k + 1) * LORA_DIM + col];   // B v1: K = k0 + kb + 1
            c = __builtin_amdgcn_wmma_f32_16x16x4_f32(
                    /*neg_a=*/false, afrag[i],
                    /*neg_b=*/false, b,
                    /*c_mod=*/(short)0, c,
                    /*reuse_a=*/false, /*reuse_b=*/false);
        }

        // D layout: VGPR r -> row (lane>=16 ? 8 : 0) + r, col = lane&15
        const int mrow = m0 + ((lane >> 4) << 3);
#pragma unroll
        for (int r = 0; r < 8; ++r) {
            out[(mrow + r) * LORA_DIM + col] = c[r];
        }
    }
}

extern "C" void kernel_launch(void* const* d_in, const int* in_sizes, int n_in,
                              void* d_out, int out_size, void* d_ws, size_t ws_size,
                              hipStream_t stream) {
    const int*   ids  = (const int*)d_in[0];    // [B*S] = 8192
    const float* down = (const float*)d_in[1];  // [32000, 16]
    const float* up   = (const float*)d_in[2];  // [16, 4096]
    float*       out  = (float*)d_out;          // [8192, 4096]

    const int M = in_sizes[0];                  // 8192 tokens
    dim3 grid(LORA_DIM / BLOCK_COLS, M / 16);   // (8, 512)
    dim3 block(256);
    lora_embed_wmma_kernel<<<grid, block, 0, stream>>>(ids, down, up, out);
}